// MAMCA_47339129537100
// MI455X (gfx1250) — compile-verified
//
#include <hip/hip_runtime.h>
#include <math.h>

// ---------------------------------------------------------------------------
// MI455X (gfx1250) implementation of the Mamba classifier reference.
// Matrix math uses V_WMMA_F32_16X16X4_F32 (fp32 in/out, wave32, 16x16 C tile).
//
// WMMA f32 16x16x4 register layouts (CDNA5 ISA 7.12.2):
//   A (16x4, MxK), 2 VGPRs: lanes 0-15: v0=(M=lane,K=0) v1=(M=lane,K=1)
//                           lanes 16-31: v0=(M=lane-16,K=2) v1=(...,K=3)
//   B (4x16, KxN), 2 VGPRs: lanes 0-15: v0=(K=0,N=lane) v1=(K=1,N=lane)
//                           lanes 16-31: v0=(K=2,N=lane-16) v1=(K=3,...)
//   C/D (16x16), 8 VGPRs:   v[g]: lanes 0-15 = (M=g, N=lane)
//                                 lanes 16-31 = (M=g+8, N=lane-16)
//
// GEMM inner loops are branch-free: out-of-range B columns are handled by
// clamping the row pointer (their products land only in C columns that are
// never stored), so EXEC stays all-ones and loads can be clause-combined.
// ---------------------------------------------------------------------------

typedef __attribute__((ext_vector_type(2))) float v2f;
typedef __attribute__((ext_vector_type(8))) float v8f;

#define B_      32
#define L_      1000
#define DM      128
#define DI      256
#define DSTATE  16
#define DTR     8
#define XDBW    40          // DT_RANK + 2*D_STATE
#define ROWS    (B_ * L_)   // 32000
#define EPS_    1e-5f
#define FC1K    (DM * L_)   // 128000
#define FC1N    256
#define KSPLIT  128         // fc1 split-K chunks (K chunk = 1000)

__device__ __forceinline__ v8f wmma_k4(v2f a, v2f b, v8f c) {
  // D = A(16x4) * B(4x16) + C, all fp32
  return __builtin_amdgcn_wmma_f32_16x16x4_f32(
      /*neg_a=*/false, a, /*neg_b=*/false, b,
      /*c_mod=*/(short)0, c, /*reuse_a=*/false, /*reuse_b=*/false);
}

__device__ __forceinline__ float silu_(float x) {
  return x / (1.f + __expf(-x));
}

// -------------------- front conv1 (1->128, k=3, pad 1) + BN + ReLU ---------
// out h1 stored channels-last: h1[(b*L + l)*128 + c]
__global__ void conv1_bn_relu(const float* __restrict__ x,
                              const float* __restrict__ w1,
                              const float* __restrict__ b1,
                              const float* __restrict__ bng,
                              const float* __restrict__ bnb,
                              const float* __restrict__ bnm,
                              const float* __restrict__ bnv,
                              float* __restrict__ h1) {
  int idx = blockIdx.x * blockDim.x + threadIdx.x;
  if (idx >= ROWS * DM) return;
  int c = idx & (DM - 1);
  int r = idx >> 7;            // b*L + l
  int b = r / L_, l = r - b * L_;
  const float* xb = x + (size_t)b * L_;
  float acc = b1[c];
  float w0 = w1[c * 3 + 0], wm = w1[c * 3 + 1], wp = w1[c * 3 + 2];
  if (l > 0)      acc += w0 * xb[l - 1];
  acc += wm * xb[l];
  if (l < L_ - 1) acc += wp * xb[l + 1];
  float scale = bng[c] * rsqrtf(bnv[c] + EPS_);
  float val = (acc - bnm[c]) * scale + bnb[c];
  h1[(size_t)r * DM + c] = fmaxf(val, 0.f);
}

// -------------------- conv2 (128->128, k=3, pad 1) as implicit WMMA GEMM ---
// grid (2000 m-tiles, 8 n-tiles), 32 threads. out stored (ROWS, 128).
// Branch-free: boundary taps use a clamped pointer and a 0/1 multiplier.
__global__ void conv2_bn_relu_wmma(const float* __restrict__ h1,
                                   const float* __restrict__ w2,   // (128,128,3)
                                   const float* __restrict__ cb,
                                   const float* __restrict__ bng,
                                   const float* __restrict__ bnb,
                                   const float* __restrict__ bnm,
                                   const float* __restrict__ bnv,
                                   float* __restrict__ out) {
  int tile_m = blockIdx.x;
  int tile_n = blockIdx.y;
  int lane = threadIdx.x;
  int half = lane >> 4, l15 = lane & 15;
  int r = tile_m * 16 + l15;
  int b = r / L_, l = r - b * L_;
  int co = tile_n * 16 + l15;
  const float* Wbase = w2 + (size_t)co * (DM * 3);
  v8f acc = {};
#pragma unroll
  for (int tap = 0; tap < 3; ++tap) {
    int ll = l + tap - 1;
    bool valid = (ll >= 0) && (ll < L_);
    float msk = valid ? 1.f : 0.f;
    int lls = valid ? ll : l;                        // safe in-range address
    const float* Arow = h1 + ((size_t)(b * L_ + lls)) * DM + half * 2;
    const float* Wr = Wbase + tap + half * 6;        // (half*2)*3
#pragma unroll 4
    for (int k = 0; k < DM; k += 4) {
      v2f a = *(const v2f*)(Arow + k);
      a.x *= msk;
      a.y *= msk;
      v2f bv;
      bv.x = Wr[k * 3];
      bv.y = Wr[k * 3 + 3];
      acc = wmma_k4(a, bv, acc);
    }
  }
  int col = tile_n * 16 + l15;
  float scale = bng[col] * rsqrtf(bnv[col] + EPS_);
  float mm = bnm[col], bb = bnb[col], cbias = cb[col];
#pragma unroll
  for (int g = 0; g < 8; ++g) {
    int row = tile_m * 16 + half * 8 + g;
    float val = (acc[g] + cbias - mm) * scale + bb;
    out[(size_t)row * DM + col] = fmaxf(val, 0.f);
  }
}

// -------------------- residual add + LayerNorm (D=128) ----------------------
// one wave per row (4 floats/lane), 8 rows per 256-thread block
__global__ void add_layernorm(const float* __restrict__ h,
                              float* __restrict__ res,
                              float* __restrict__ out,
                              const float* __restrict__ w,
                              const float* __restrict__ b,
                              int add) {
  int wid = threadIdx.x >> 5;
  int lane = threadIdx.x & 31;
  int row = blockIdx.x * 8 + wid;
  if (row >= ROWS) return;
  const float4* hr = (const float4*)(h + (size_t)row * DM);
  float4* rr = (float4*)(res + (size_t)row * DM);
  float4 hv = hr[lane];
  if (add) {
    float4 rv = rr[lane];
    hv.x += rv.x; hv.y += rv.y; hv.z += rv.z; hv.w += rv.w;
  }
  rr[lane] = hv;
  float s  = hv.x + hv.y + hv.z + hv.w;
  float s2 = hv.x*hv.x + hv.y*hv.y + hv.z*hv.z + hv.w*hv.w;
  for (int off = 16; off > 0; off >>= 1) {
    s  += __shfl_xor(s,  off, 32);
    s2 += __shfl_xor(s2, off, 32);
  }
  float mu  = s * (1.f / DM);
  float var = s2 * (1.f / DM) - mu * mu;
  float inv = rsqrtf(var + EPS_);
  float4 wv = ((const float4*)w)[lane];
  float4 bv = ((const float4*)b)[lane];
  float4 o;
  o.x = (hv.x - mu) * inv * wv.x + bv.x;
  o.y = (hv.y - mu) * inv * wv.y + bv.y;
  o.z = (hv.z - mu) * inv * wv.z + bv.z;
  o.w = (hv.w - mu) * inv * wv.w + bv.w;
  ((float4*)(out + (size_t)row * DM))[lane] = o;
}

// -------------------- generic WMMA GEMM: C(MxN) = A(MxK) * W(NxK)^T --------
// One wave computes a 16 x (16*NT) strip: one A fetch feeds NT WMMAs.
// grid = (M/16, ceil(N/(16*NT))). M must be a multiple of 16.
// NG: N not a multiple of 16*NT -> clamp W rows for loads, guard stores.
template <int K, int NT, bool NG>
__global__ void gemm_wmma_nt(const float* __restrict__ A,
                             const float* __restrict__ W,
                             float* __restrict__ C,
                             int N, int ldc) {
  int lane = threadIdx.x;
  int half = lane >> 4, l15 = lane & 15;
  const float* Arow = A + (size_t)(blockIdx.x * 16 + l15) * K + half * 2;
  int nbase = blockIdx.y * (16 * NT);
  const float* Wrow[NT];
  v8f acc[NT] = {};
#pragma unroll
  for (int t = 0; t < NT; ++t) {
    int bn = nbase + t * 16 + l15;
    if (NG && bn >= N) bn = N - 1;       // clamp: products land in unstored cols
    Wrow[t] = W + (size_t)bn * K + half * 2;
  }
#pragma unroll 4
  for (int k = 0; k < K; k += 4) {
    v2f a = *(const v2f*)(Arow + k);
#pragma unroll
    for (int t = 0; t < NT; ++t) {
      v2f b = *(const v2f*)(Wrow[t] + k);
      acc[t] = wmma_k4(a, b, acc[t]);
    }
  }
  int row0 = blockIdx.x * 16 + half * 8;
#pragma unroll
  for (int t = 0; t < NT; ++t) {
    int col = nbase + t * 16 + l15;
    if (!NG || col < N) {
#pragma unroll
      for (int g = 0; g < 8; ++g)
        C[(size_t)(row0 + g) * ldc + col] = acc[t][g];
    }
  }
}

// -------------------- causal depthwise conv (k=4) + SiLU --------------------
// x part of xz: xz[r*512 + d], d in [0,256)
__global__ void dwconv_silu(const float* __restrict__ xz,
                            const float* __restrict__ cw,   // (256,1,4)
                            const float* __restrict__ cb,
                            float* __restrict__ xs) {
  int idx = blockIdx.x * blockDim.x + threadIdx.x;
  if (idx >= ROWS * DI) return;
  int d = idx & (DI - 1);
  int r = idx >> 8;
  int b = r / L_, l = r - b * L_;
  const float* w = cw + d * 4;
  float acc = cb[d];
#pragma unroll
  for (int k = 0; k < 4; ++k) {
    int ll = l - 3 + k;
    if (ll >= 0) acc += w[k] * xz[((size_t)(b * L_ + ll)) * (2 * DI) + d];
  }
  xs[(size_t)r * DI + d] = silu_(acc);
}

// -------------------- dt = softplus(xdb[:, :8] @ dt_w^T + dt_b) -------------
__global__ void dt_softplus(const float* __restrict__ xdb,
                            const float* __restrict__ dtw,  // (256,8)
                            const float* __restrict__ dtb,
                            float* __restrict__ dt) {
  int idx = blockIdx.x * blockDim.x + threadIdx.x;
  if (idx >= ROWS * DI) return;
  int d = idx & (DI - 1);
  int r = idx >> 8;
  const float* xr = xdb + (size_t)r * XDBW;
  const float* wr = dtw + d * DTR;
  float acc = dtb[d];
#pragma unroll
  for (int j = 0; j < DTR; ++j) acc += wr[j] * xr[j];
  float sp = (acc > 20.f) ? acc : __logf(1.f + __expf(acc));
  dt[(size_t)r * DI + d] = sp;
}

// -------------------- selective scan (serial over L) + D skip + SiLU gate ---
// thread = (b, d, s); 16-lane shfl_xor reduction over state dim per step.
// block = 256 threads (16 d's), grid = B_ * (DI/16) = 512 blocks.
__global__ void selective_scan(const float* __restrict__ xs,
                               const float* __restrict__ dt,
                               const float* __restrict__ xdb,
                               const float* __restrict__ xz,
                               const float* __restrict__ A_log,
                               const float* __restrict__ Dp,
                               float* __restrict__ y) {
  int b  = blockIdx.x >> 4;
  int dg = blockIdx.x & 15;
  int t  = threadIdx.x;
  int dl = t >> 4;
  int s  = t & 15;
  int d  = dg * 16 + dl;
  float A  = -__expf(A_log[d * DSTATE + s]);
  float Dv = Dp[d];
  float h  = 0.f;
  size_t rbase = (size_t)b * L_;
  for (int l = 0; l < L_; ++l) {
    size_t r = rbase + l;
    float xv  = xs[r * DI + d];
    float dtv = dt[r * DI + d];
    float Bv  = xdb[r * XDBW + DTR + s];
    float Cv  = xdb[r * XDBW + DTR + DSTATE + s];
    float dA  = __expf(dtv * A);
    h = dA * h + (dtv * xv) * Bv;
    float yc = h * Cv;
#pragma unroll
    for (int off = 8; off > 0; off >>= 1)
      yc += __shfl_xor(yc, off, 32);       // reduces within each 16-lane group
    if (s == 0) {
      float z  = xz[r * (2 * DI) + DI + d];
      float yv = (yc + xv * Dv) * silu_(z);
      y[r * DI + d] = yv;
    }
  }
}

// -------------------- fc1: split-K WMMA, deterministic partials -------------
// grid (2 mtiles * 4 ngroups, 128 chunks); K chunk = 1000 (250 steps x 4 wmma)
__global__ void fc1_wmma_split(const float* __restrict__ hf,   // (32, 128000)
                               const float* __restrict__ w,    // (256, 128000)
                               float* __restrict__ part) {     // (128, 32, 256)
  int tile_m = blockIdx.x >> 2;
  int ng = blockIdx.x & 3;
  int chunk = blockIdx.y;
  int lane = threadIdx.x;
  int half = lane >> 4, l15 = lane & 15;
  const int KC = FC1K / KSPLIT;   // 1000
  const float* Arow = hf + (size_t)(tile_m * 16 + l15) * FC1K + chunk * KC + half * 2;
  const float* Wrow[4];
  v8f acc[4] = {};
#pragma unroll
  for (int t = 0; t < 4; ++t)
    Wrow[t] = w + (size_t)(ng * 64 + t * 16 + l15) * FC1K + chunk * KC + half * 2;
#pragma unroll 4
  for (int k = 0; k < KC; k += 4) {
    __builtin_prefetch(Wrow[0] + k + 512, 0, 0);   // stream weights ahead
    v2f a = *(const v2f*)(Arow + k);
#pragma unroll
    for (int t = 0; t < 4; ++t) {
      v2f b = *(const v2f*)(Wrow[t] + k);
      acc[t] = wmma_k4(a, b, acc[t]);
    }
  }
  float* pc = part + (size_t)chunk * (B_ * FC1N);
  int row0 = tile_m * 16 + half * 8;
#pragma unroll
  for (int t = 0; t < 4; ++t) {
    int col = ng * 64 + t * 16 + l15;
#pragma unroll
    for (int g = 0; g < 8; ++g)
      pc[(row0 + g) * FC1N + col] = acc[t][g];
  }
}

__global__ void fc1_reduce_bn_relu(const float* __restrict__ part,
                                   const float* __restrict__ fb,
                                   const float* __restrict__ bng,
                                   const float* __restrict__ bnb,
                                   const float* __restrict__ bnm,
                                   const float* __restrict__ bnv,
                                   float* __restrict__ a1) {
  int idx = blockIdx.x * blockDim.x + threadIdx.x;
  if (idx >= B_ * FC1N) return;
  float s = 0.f;
  for (int c = 0; c < KSPLIT; ++c) s += part[(size_t)c * (B_ * FC1N) + idx];
  int n = idx & (FC1N - 1);
  s += fb[n];
  float scale = bng[n] * rsqrtf(bnv[n] + EPS_);
  float val = (s - bnm[n]) * scale + bnb[n];
  a1[idx] = fmaxf(val, 0.f);
}

// -------------------- fc2 (256 -> 2) ---------------------------------------
__global__ void fc2_kernel(const float* __restrict__ a1,
                           const float* __restrict__ w,   // (2,256)
                           const float* __restrict__ b,
                           float* __restrict__ out) {
  int idx = threadIdx.x;
  if (idx >= B_ * 2) return;
  int m = idx >> 1, n = idx & 1;
  const float* ar = a1 + m * FC1N;
  const float* wr = w + n * FC1N;
  float s = b[n];
  for (int k = 0; k < FC1N; ++k) s += ar[k] * wr[k];
  out[m * 2 + n] = s;
}

// ---------------------------------------------------------------------------
// Input index map (jax pytree flatten: dict keys alphabetical, lists in order):
//  0 hidden_states
//  1..4   cbn  {b,g,m,v}
//  5..8   dbn1 {b,g,m,v}
//  9..12  dbn2 {b,g,m,v}
//  13 dconv1_b  14 dconv1_w  15 dconv2_b  16 dconv2_w
//  17 fc1_b  18 fc1_w  19 fc2_b  20 fc2_w
//  21 + 11*i per layer i: A_log, D, conv_b, conv_w, dt_proj_b, dt_proj_w,
//                         in_proj_w, norm_b, norm_w, out_proj_w, x_proj_w
//  65 norm_f_b  66 norm_f_w
// ---------------------------------------------------------------------------

// workspace layout (float offsets)
#define OFF_H1   ((size_t)0)                      // (ROWS,128)
#define OFF_H    (OFF_H1  + (size_t)ROWS * DM)    // (ROWS,128)
#define OFF_RES  (OFF_H   + (size_t)ROWS * DM)    // (ROWS,128)
#define OFF_HN   (OFF_RES + (size_t)ROWS * DM)    // (ROWS,128)  also final hf
#define OFF_XZ   (OFF_HN  + (size_t)ROWS * DM)    // (ROWS,512)  reused: fc1 partials
#define OFF_XS   (OFF_XZ  + (size_t)ROWS * 2*DI)  // (ROWS,256)  reused: a1
#define OFF_XDB  (OFF_XS  + (size_t)ROWS * DI)    // (ROWS,40)
#define OFF_DT   (OFF_XDB + (size_t)ROWS * XDBW)  // (ROWS,256)
#define OFF_Y    (OFF_DT  + (size_t)ROWS * DI)    // (ROWS,256)

extern "C" void kernel_launch(void* const* d_in, const int* in_sizes, int n_in,
                              void* d_out, int out_size, void* d_ws, size_t ws_size,
                              hipStream_t stream) {
  const float* x = (const float*)d_in[0];
  float* ws = (float*)d_ws;
  float* h1  = ws + OFF_H1;
  float* h   = ws + OFF_H;
  float* res = ws + OFF_RES;
  float* hn  = ws + OFF_HN;
  float* xz  = ws + OFF_XZ;
  float* xs  = ws + OFF_XS;
  float* xdb = ws + OFF_XDB;
  float* dt  = ws + OFF_DT;
  float* y   = ws + OFF_Y;
  float* part = ws + OFF_XZ;   // fc1 partials reuse xz region
  float* a1   = ws + OFF_XS;   // fc1 output reuses xs region

  // front convs
  conv1_bn_relu<<<(ROWS * DM + 255) / 256, 256, 0, stream>>>(
      x, (const float*)d_in[14], (const float*)d_in[13],
      (const float*)d_in[6], (const float*)d_in[5],
      (const float*)d_in[7], (const float*)d_in[8], h1);
  conv2_bn_relu_wmma<<<dim3(ROWS / 16, DM / 16), 32, 0, stream>>>(
      h1, (const float*)d_in[16], (const float*)d_in[15],
      (const float*)d_in[10], (const float*)d_in[9],
      (const float*)d_in[11], (const float*)d_in[12], h);

  // mamba layers
  for (int i = 0; i < 4; ++i) {
    const int base = 21 + 11 * i;
    const float* A_log = (const float*)d_in[base + 0];
    const float* Dp    = (const float*)d_in[base + 1];
    const float* cvb   = (const float*)d_in[base + 2];
    const float* cvw   = (const float*)d_in[base + 3];
    const float* dtb   = (const float*)d_in[base + 4];
    const float* dtw   = (const float*)d_in[base + 5];
    const float* inw   = (const float*)d_in[base + 6];
    const float* nb    = (const float*)d_in[base + 7];
    const float* nw    = (const float*)d_in[base + 8];
    const float* outw  = (const float*)d_in[base + 9];
    const float* xpw   = (const float*)d_in[base + 10];

    add_layernorm<<<ROWS / 8, 256, 0, stream>>>(h, res, hn, nw, nb, i > 0 ? 1 : 0);
    // in_proj: (32000x128) @ (128->512)
    gemm_wmma_nt<DM, 4, false><<<dim3(ROWS / 16, (2 * DI) / 64), 32, 0, stream>>>(
        hn, inw, xz, 2 * DI, 2 * DI);
    dwconv_silu<<<(ROWS * DI + 255) / 256, 256, 0, stream>>>(xz, cvw, cvb, xs);
    // x_proj: (32000x256) @ (256->40), 3 N-tiles with store guard
    gemm_wmma_nt<DI, 3, true><<<dim3(ROWS / 16, 1), 32, 0, stream>>>(
        xs, xpw, xdb, XDBW, XDBW);
    dt_softplus<<<(ROWS * DI + 255) / 256, 256, 0, stream>>>(xdb, dtw, dtb, dt);
    selective_scan<<<B_ * (DI / 16), 256, 0, stream>>>(xs, dt, xdb, xz, A_log, Dp, y);
    // out_proj: (32000x256) @ (256->128)
    gemm_wmma_nt<DI, 4, false><<<dim3(ROWS / 16, DM / 64), 32, 0, stream>>>(
        y, outw, h, DM, DM);
  }

  // final residual + LN -> hf (flattened (32, 128000))
  add_layernorm<<<ROWS / 8, 256, 0, stream>>>(
      h, res, hn, (const float*)d_in[66], (const float*)d_in[65], 1);

  // classifier head
  fc1_wmma_split<<<dim3((B_ / 16) * (FC1N / 64), KSPLIT), 32, 0, stream>>>(
      hn, (const float*)d_in[18], part);
  fc1_reduce_bn_relu<<<(B_ * FC1N + 255) / 256, 256, 0, stream>>>(
      part, (const float*)d_in[17],
      (const float*)d_in[2], (const float*)d_in[1],
      (const float*)d_in[3], (const float*)d_in[4], a1);
  fc2_kernel<<<1, 64, 0, stream>>>(
      a1, (const float*)d_in[20], (const float*)d_in[19], (float*)d_out);
}